// LDSVariational_52432960749772
// MI455X (gfx1250) — compile-verified
//
#include <hip/hip_runtime.h>

#define TT 256
#define DD 32
#define KK 8
#define NB 64
#define AUG_LD 66
#define EPS_TRI 0.01f
#define EPS_CHOL 0.01f
#define EPS_BS 0.0001f

typedef float v2f __attribute__((ext_vector_type(2)));
typedef float v8f __attribute__((ext_vector_type(8)));

enum { MM_STORE = 0, MM_STORE_NEG = 1, MM_SUB = 2, MM_ADD = 3 };

// 32x32 matmul D(op)= opA(A) * opB(B) using V_WMMA_F32_16X16X4_F32.
// 4 waves: wave w computes 16x16 tile (w>>1, w&1). A,B in LDS row-major ld=32.
template <bool TA, bool TB>
__device__ inline void mm32(const float* A, const float* B, float* D, int ldd,
                            int mode, int tid) {
  const int lane = tid & 31;
  const int wid = tid >> 5;
  const int mt = (wid >> 1) << 4;
  const int nt = (wid & 1) << 4;
  const int l15 = lane & 15;
  const int kh = (lane >> 4) << 1;  // 0 or 2
  v8f acc = {};
#pragma unroll
  for (int kt = 0; kt < 32; kt += 4) {
    const int k0 = kt + kh;
    v2f av, bv;
    if (!TA) { av.x = A[(mt + l15) * 32 + k0];  av.y = A[(mt + l15) * 32 + k0 + 1]; }
    else     { av.x = A[k0 * 32 + mt + l15];    av.y = A[(k0 + 1) * 32 + mt + l15]; }
    if (!TB) { bv.x = B[k0 * 32 + nt + l15];    bv.y = B[(k0 + 1) * 32 + nt + l15]; }
    else     { bv.x = B[(nt + l15) * 32 + k0];  bv.y = B[(nt + l15) * 32 + k0 + 1]; }
    acc = __builtin_amdgcn_wmma_f32_16x16x4_f32(false, av, false, bv, (short)0,
                                                acc, false, false);
  }
  const int rbase = mt + ((lane >> 4) << 3);
#pragma unroll
  for (int r = 0; r < 8; ++r) {
    float v = acc[r];
    float* p = &D[(rbase + r) * ldd + nt + l15];
    if (mode == MM_STORE) *p = v;
    else if (mode == MM_STORE_NEG) *p = -v;
    else if (mode == MM_SUB) *p -= v;
    else *p += v;
  }
}

// ---------------- kernel 1: Ris[t] = R[t]*R[t]^T, hR[t] = Ris[t]*ms[t] ------
__global__ __launch_bounds__(128) void ris_precompute_kernel(
    const float* __restrict__ Ri_sqrts, const float* __restrict__ ms,
    float* __restrict__ g_Ris, float* __restrict__ g_hR) {
  __shared__ float sR[1024], sO[1024], sm[DD];
  const int t = blockIdx.x;
  const int tid = threadIdx.x;
  for (int e = tid; e < 1024; e += 128) sR[e] = Ri_sqrts[(size_t)t * 1024 + e];
  if (tid < DD) sm[tid] = ms[t * DD + tid];
  __syncthreads();
  mm32<false, true>(sR, sR, sO, 32, MM_STORE, tid);
  __syncthreads();
  for (int e = tid; e < 1024; e += 128) g_Ris[(size_t)t * 1024 + e] = sO[e];
  if (tid < DD) {
    float a = 0.f;
    for (int j = 0; j < DD; ++j) a += sO[tid * 32 + j] * sm[j];
    g_hR[t * DD + tid] = a;
  }
}

// ---------------- kernel 2: fused forward pass (tridiag solve + chol + samp) -
__global__ __launch_bounds__(128) void lds_forward_kernel(
    const float* __restrict__ z_samples, const float* __restrict__ A_base,
    const float* __restrict__ b_base, const float* __restrict__ Q_sqrt,
    const float* __restrict__ noise, const float* __restrict__ g_Ris,
    const float* __restrict__ g_hR, float* __restrict__ g_c,
    float* __restrict__ g_d, float* __restrict__ out) {
  __shared__ float sA[1024], sQs[1024], sJd[1024], sC[1024], sL[1024],
      sLl[1024], sM[1024];
  __shared__ float sQis[2][1024], sJl[2][1024];
  __shared__ float sAug[32 * AUG_LD];
  __shared__ float sz[KK], sbs[DD], sQb[DD], sh[DD], sJlPd[DD], sd[DD],
      srhs[DD], sx[DD];

  const int n = blockIdx.x;
  const int tid = threadIdx.x;
  const int row = tid & 31;
  const int jgrp = tid >> 5;

  for (int t = 0; t < TT; ++t) {
    const int cur = t & 1, prev = cur ^ 1;
    const bool haveNext = (t < TT - 1);

    if (haveNext) {
      if (tid < KK) sz[tid] = z_samples[((size_t)n * TT + t) * KK + tid];
      __syncthreads();
      for (int e = tid; e < 1024; e += 128) {  // mix As, Qs
        float a = 0.f, q = 0.f;
#pragma unroll
        for (int k = 0; k < KK; ++k) {
          float zk = sz[k];
          a += zk * A_base[k * 1024 + e];
          q += zk * Q_sqrt[k * 1024 + e];
        }
        sA[e] = a;
        sQs[e] = q;
      }
      if (tid < DD) {  // mix bs
        float b = 0.f;
#pragma unroll
        for (int k = 0; k < KK; ++k) b += sz[k] * b_base[k * DD + tid];
        sbs[tid] = b;
      }
      __syncthreads();
      mm32<false, true>(sQs, sQs, sQis[cur], 32, MM_STORE, tid);  // Qis=Qs*Qs^T
      __syncthreads();
      mm32<false, false>(sQis[cur], sA, sJl[cur], 32, MM_STORE_NEG, tid);  // Jl
      __syncthreads();
      mm32<true, false>(sA, sJl[cur], sJd, 32, MM_STORE_NEG, tid);  // -As^T*Jl
      __syncthreads();
    } else {
      for (int e = tid; e < 1024; e += 128) sJd[e] = 0.f;
      __syncthreads();
    }

    // finalize Jd; h_t; Jl_prev*d_prev; next Qb = Qis_t*bs_t
    for (int e = tid; e < 1024; e += 128) {
      float v = sJd[e] + g_Ris[(size_t)t * 1024 + e];
      if (t > 0) v += sQis[prev][e];
      sJd[e] = v;
    }
    if (tid < DD) {
      float hv = g_hR[t * DD + tid];
      if (t > 0) hv += sQb[tid];
      if (haveNext) {
        float jb = 0.f;
        for (int j = 0; j < DD; ++j) jb += sJl[cur][tid * 32 + j] * sbs[j];
        hv += jb;
      }
      float jd = 0.f;
      if (t > 0)
        for (int j = 0; j < DD; ++j) jd += sJl[prev][tid * 32 + j] * sd[j];
      sh[tid] = hv;
      sJlPd[tid] = jd;
      if (haveNext) {
        float qb = 0.f;
        for (int j = 0; j < DD; ++j) qb += sQis[cur][tid * 32 + j] * sbs[j];
        sQb[tid] = qb;
      }
    }
    __syncthreads();

    // augmented system [ J | Jl_t^T | rhs ]
    for (int idx = tid; idx < 32 * 65; idx += 128) {
      int i = idx / 65, j = idx % 65;
      float v;
      if (j < 32)      v = sJd[i * 32 + j] + (i == j ? EPS_TRI : 0.f);
      else if (j < 64) v = haveNext ? sJl[cur][(j - 32) * 32 + i] : 0.f;
      else             v = sh[i] - (t > 0 ? sJlPd[i] : 0.f);
      sAug[i * AUG_LD + j] = v;
    }
    __syncthreads();
    if (t > 0) {  // J -= Jl_prev * c_prev
      mm32<false, false>(sJl[prev], sC, sAug, AUG_LD, MM_SUB, tid);
      __syncthreads();
    }
    // Gauss-Jordan (no pivoting; J is SPD Schur complement)
    for (int k = 0; k < 32; ++k) {
      float piv = sAug[k * AUG_LD + k];
      float f = sAug[row * AUG_LD + k] / piv;
      __syncthreads();
      if (row != k) {
        for (int j = k + 1 + jgrp; j < 65; j += 4)
          sAug[row * AUG_LD + j] -= f * sAug[k * AUG_LD + j];
        if (jgrp == 0) sAug[row * AUG_LD + k] = 0.f;
      }
      __syncthreads();
    }
    // extract c_t = (J^-1 Jl_t^T)^T, d_t
    for (int e = tid; e < 1024; e += 128) {
      int i = e >> 5, j = e & 31;
      sC[e] = sAug[j * AUG_LD + 32 + i] / sAug[j * AUG_LD + j];
    }
    if (tid < DD) sd[tid] = sAug[tid * AUG_LD + 64] / sAug[tid * AUG_LD + tid];
    __syncthreads();
    {
      size_t base = (size_t)n * TT + t;
      if (haveNext)
        for (int e = tid; e < 1024; e += 128) g_c[base * 1024 + e] = sC[e];
      if (tid < DD) g_d[base * DD + tid] = sd[tid];
    }

    // ---- Cholesky chain ----
    if (t > 0) {
      // Ll = Jl_prev * L_prev^{-1}: solve L_prev^T X = Jl_prev^T per column
      for (int e = tid; e < 1024; e += 128)
        sAug[(e >> 5) * AUG_LD + (e & 31)] = sJl[prev][e];
      __syncthreads();
      if (tid < 32) {
        int c = tid;
        for (int i = 31; i >= 0; --i) {
          float xi = sAug[c * AUG_LD + i] / sL[i * 32 + i];
          sAug[c * AUG_LD + i] = xi;
          for (int j = 0; j < i; ++j)
            sAug[c * AUG_LD + j] -= sL[i * 32 + j] * xi;
        }
      }
      __syncthreads();
      for (int e = tid; e < 1024; e += 128)
        sLl[e] = sAug[(e >> 5) * AUG_LD + (e & 31)];
      __syncthreads();
      for (int e = tid; e < 1024; e += 128) {
        int i = e >> 5, j = e & 31;
        sM[e] = sJd[e] + (i == j ? EPS_CHOL : 0.f);
      }
      __syncthreads();
      mm32<false, true>(sLl, sLl, sM, 32, MM_SUB, tid);  // M -= Ll*Ll^T
      __syncthreads();
    } else {
      for (int e = tid; e < 1024; e += 128) {
        int i = e >> 5, j = e & 31;
        sM[e] = sJd[e] + (i == j ? EPS_CHOL : 0.f);
      }
      __syncthreads();
    }
    // in-place Cholesky of sM (kept full-symmetric)
    for (int k = 0; k < 32; ++k) {
      float akk = sM[k * 32 + k];
      __syncthreads();
      float lkk = sqrtf(akk);
      if (tid == k) sM[k * 32 + k] = lkk;
      else if (tid < 32 && tid > k) sM[tid * 32 + k] /= lkk;
      __syncthreads();
      if (row > k)
        for (int j = k + 1 + jgrp; j < 32; j += 4)
          sM[row * 32 + j] -= sM[row * 32 + k] * sM[j * 32 + k];
      __syncthreads();
    }
    for (int e = tid; e < 1024; e += 128) {
      int i = e >> 5, j = e & 31;
      sL[e] = (j <= i) ? sM[e] : 0.f;
    }
    __syncthreads();

    // ---- sampling: (L+eps_bs I)^T x = z - Ll^T x_prev ----
    if (tid < DD) {
      float r = noise[((size_t)n * TT + t) * DD + tid];
      if (t > 0) {
        float s = 0.f;
        for (int j = 0; j < DD; ++j) s += sLl[j * 32 + tid] * sx[j];
        r -= s;
      }
      srhs[tid] = r;
    }
    __syncthreads();
    for (int i = 31; i >= 0; --i) {
      if (tid == 0) srhs[i] /= (sL[i * 32 + i] + EPS_BS);
      __syncthreads();
      if (tid < i) srhs[tid] -= sL[i * 32 + tid] * srhs[i];
      __syncthreads();
    }
    if (tid < DD) {
      sx[tid] = srhs[tid];
      out[((size_t)n * TT + t) * DD + tid] = srhs[tid];
    }
    __syncthreads();
  }
}

// ---------------- kernel 3: backward recursion mu_t = d_t - c_t*mu_{t+1} ----
__global__ __launch_bounds__(128) void lds_backward_kernel(
    const float* __restrict__ g_c, const float* __restrict__ g_d,
    float* __restrict__ out) {
  __shared__ float sCt[1024];
  __shared__ float sMu[DD], sNew[DD];
  const int n = blockIdx.x;
  const int tid = threadIdx.x;
  if (tid < DD) {
    float v = g_d[((size_t)n * TT + (TT - 1)) * DD + tid];
    sMu[tid] = v;
    out[((size_t)n * TT + (TT - 1)) * DD + tid] += v;
  }
  __syncthreads();
  for (int t = TT - 2; t >= 0; --t) {
    size_t base = (size_t)n * TT + t;
    for (int e = tid; e < 1024; e += 128) sCt[e] = g_c[base * 1024 + e];
    __syncthreads();
    if (tid < DD) {
      float acc = g_d[base * DD + tid];
      for (int j = 0; j < DD; ++j) acc -= sCt[tid * 32 + j] * sMu[j];
      sNew[tid] = acc;
    }
    __syncthreads();
    if (tid < DD) {
      sMu[tid] = sNew[tid];
      out[base * DD + tid] += sNew[tid];
    }
    __syncthreads();
  }
}

extern "C" void kernel_launch(void* const* d_in, const int* in_sizes, int n_in,
                              void* d_out, int out_size, void* d_ws,
                              size_t ws_size, hipStream_t stream) {
  (void)in_sizes; (void)n_in; (void)out_size; (void)ws_size;
  const float* z_samples = (const float*)d_in[0];
  const float* A_base = (const float*)d_in[1];
  const float* b_base = (const float*)d_in[2];
  const float* Q_sqrt = (const float*)d_in[3];
  const float* ms = (const float*)d_in[4];
  const float* Ri_sqrts = (const float*)d_in[5];
  const float* noise = (const float*)d_in[6];
  float* ws = (float*)d_ws;
  float* g_Ris = ws;                               // T*D*D
  float* g_hR = g_Ris + (size_t)TT * DD * DD;      // T*D
  float* g_d = g_hR + (size_t)TT * DD;             // n*T*D
  float* g_c = g_d + (size_t)NB * TT * DD;         // n*T*D*D
  float* out = (float*)d_out;

  ris_precompute_kernel<<<TT, 128, 0, stream>>>(Ri_sqrts, ms, g_Ris, g_hR);
  lds_forward_kernel<<<NB, 128, 0, stream>>>(z_samples, A_base, b_base, Q_sqrt,
                                             noise, g_Ris, g_hR, g_c, g_d, out);
  lds_backward_kernel<<<NB, 128, 0, stream>>>(g_c, g_d, out);
}